// geoparse_model_20066087207045
// MI455X (gfx1250) — compile-verified
//
#include <hip/hip_runtime.h>
#include <math.h>

#define EPSF 1e-8f

typedef __attribute__((ext_vector_type(2))) float v2f;
typedef __attribute__((ext_vector_type(4))) float v4f;
typedef __attribute__((ext_vector_type(8))) float v8f;

#define HAVE_WMMA_F32X4 __has_builtin(__builtin_amdgcn_wmma_f32_16x16x4_f32)

// ---------------------------------------------------------------------------
// Kernel A: all BERT->small projections via V_WMMA_F32_16X16X4_F32.
//   job 0: placename[2048,768] x [W_t2c(24);W_code(8)]^T -> xall[2048][32]
//   job 1: other_locs          x W_ctx^T (24, pad 32)    -> xoth[2048][32]
//   job 2: doc                 x W_ctx^T                 -> xdoc[2048][32]
//   job 3: country_emb[250,768]x W_cet^T (hoisted gather)-> cetp[256][32]
//
// 256-thread block (8 waves) per 16-row M tile. K=768 is split 96-per-wave
// (short dependency chains, 3200 waves total); partial accumulators are
// reduced through LDS. Each 8-wide K chunk is fed by one b128 load per
// operand: half-wave 0 takes k[B..B+4), half-wave 1 takes k[B+4..B+8) --
// a pure permutation of the WMMA K slots, applied identically to A and B,
// so the fp32 dot product is exact. 12 b128 loads are batched per round to
// keep memory latency hidden.
// ---------------------------------------------------------------------------
__global__ __launch_bounds__(256)
void proj_wmma_kernel(const float* __restrict__ placename,
                      const float* __restrict__ other,
                      const float* __restrict__ doc,
                      const float* __restrict__ cemb,
                      const float* __restrict__ W_t2c, const float* __restrict__ b_t2c,
                      const float* __restrict__ W_code, const float* __restrict__ b_code,
                      const float* __restrict__ W_ctx, const float* __restrict__ b_ctx,
                      const float* __restrict__ W_cet, const float* __restrict__ b_cet,
                      float* __restrict__ xall, float* __restrict__ xoth,
                      float* __restrict__ xdoc, float* __restrict__ cetp)
{
    __shared__ float sacc[8 * 512];   // [wave][tile(2)][j(8)][lane(32)] = 16 KB

    int bid = blockIdx.x;
    int job, mtile;
    if (bid < 128)      { job = 0; mtile = bid; }
    else if (bid < 256) { job = 1; mtile = bid - 128; }
    else if (bid < 384) { job = 2; mtile = bid - 256; }
    else                { job = 3; mtile = bid - 384; }

    const float* A; const float* W; const float* bias; float* out; int mMax;
    switch (job) {
      case 0:  A = placename; W = W_t2c; bias = b_t2c; out = xall; mMax = 2048; break;
      case 1:  A = other;     W = W_ctx; bias = b_ctx; out = xoth; mMax = 2048; break;
      case 2:  A = doc;       W = W_ctx; bias = b_ctx; out = xdoc; mMax = 2048; break;
      default: A = cemb;      W = W_cet; bias = b_cet; out = cetp; mMax = 250;  break;
    }

    const int tid   = threadIdx.x;
    const int wave  = tid >> 5;      // 0..7 : K-split
    const int lane  = tid & 31;
    const int halfM = lane >> 4;
    const int lid   = lane & 15;

    // A row (clamped for the 250-row country job; clamped rows never stored)
    int arow = mtile * 16 + lid;
    if (arow >= mMax) arow = mMax - 1;
    const float* abase = A + (size_t)arow * 768;

    // B operand: B[k][n] = W[n][k]. Per-lane row pointers for the two N tiles.
    const int col0 = lid;
    const int col1 = 16 + lid;
    const float* rp0;
    const float* rp1;
    float s1;                 // zeroes padded columns (24..31) branchlessly
    if (job == 0) {
        rp0 = W_t2c + col0 * 768;
        rp1 = (col1 < 24) ? (W_t2c + col1 * 768) : (W_code + (col1 - 24) * 768);
        s1 = 1.0f;
    } else {
        rp0 = W + col0 * 768;
        if (col1 < 24) { rp1 = W + col1 * 768; s1 = 1.0f; }
        else           { rp1 = W;              s1 = 0.0f; }
    }

    const int kbase = wave * 96;     // this wave's 96-wide K slice (12 chunks of 8)
    v8f acc0 = {}; v8f acc1 = {};

#if HAVE_WMMA_F32X4
    const int koff = kbase + halfM * 4;   // per-lane start inside each chunk
    for (int cb = 0; cb < 12; cb += 4) {
        v4f av[4], b0v[4], b1v[4];
        #pragma unroll
        for (int c = 0; c < 4; ++c) {
            int off = koff + (cb + c) * 8;
            av[c]  = *(const v4f*)(abase + off);
            b0v[c] = *(const v4f*)(rp0 + off);
            b1v[c] = *(const v4f*)(rp1 + off) * s1;
        }
        #pragma unroll
        for (int c = 0; c < 4; ++c) {
            v2f aL  = { av[c][0],  av[c][1]  };
            v2f aH  = { av[c][2],  av[c][3]  };
            v2f b0L = { b0v[c][0], b0v[c][1] };
            v2f b0H = { b0v[c][2], b0v[c][3] };
            v2f b1L = { b1v[c][0], b1v[c][1] };
            v2f b1H = { b1v[c][2], b1v[c][3] };
            acc0 = __builtin_amdgcn_wmma_f32_16x16x4_f32(false, aL, false, b0L,
                                                         (short)0, acc0, false, false);
            acc1 = __builtin_amdgcn_wmma_f32_16x16x4_f32(false, aL, false, b1L,
                                                         (short)0, acc1, false, false);
            acc0 = __builtin_amdgcn_wmma_f32_16x16x4_f32(false, aH, false, b0H,
                                                         (short)0, acc0, false, false);
            acc1 = __builtin_amdgcn_wmma_f32_16x16x4_f32(false, aH, false, b1H,
                                                         (short)0, acc1, false, false);
        }
    }
#else
    // Semantically identical scalar fallback (same D-element mapping).
    for (int j = 0; j < 8; ++j) {
        int row = mtile * 16 + j + halfM * 8;
        int r = row < mMax ? row : mMax - 1;
        const float* ar = A + (size_t)r * 768;
        float a0 = 0.f, a1 = 0.f;
        for (int k = kbase; k < kbase + 96; ++k) {
            float av = ar[k]; a0 += av * rp0[k]; a1 += av * rp1[k];
        }
        acc0[j] = a0; acc1[j] = a1 * s1;
    }
#endif

    // Publish partial accumulators, reduce across the 8 K-slices, add bias.
    #pragma unroll
    for (int j = 0; j < 8; ++j) {
        sacc[wave * 512 +       j * 32 + lane] = acc0[j];
        sacc[wave * 512 + 256 + j * 32 + lane] = acc1[j];
    }
    __syncthreads();

    {
        const int j   = tid >> 5;       // 0..7
        const int rl  = tid & 31;
        const int rh  = rl >> 4;
        const int rc  = rl & 15;
        const int row = mtile * 16 + j + 8 * rh;
        const int c0  = rc;
        const int c1  = 16 + rc;
        float v0 = 0.f, v1 = 0.f;
        #pragma unroll
        for (int w = 0; w < 8; ++w) {
            v0 += sacc[w * 512 +       j * 32 + rl];
            v1 += sacc[w * 512 + 256 + j * 32 + rl];
        }
        float bb0, bb1;
        if (job == 0) {
            bb0 = b_t2c[c0];
            bb1 = (c1 < 24) ? b_t2c[c1] : b_code[c1 - 24];
        } else {
            bb0 = bias[c0];
            bb1 = (c1 < 24) ? bias[c1] : 0.0f;
        }
        if (row < mMax) {
            out[row * 32 + c0] = v0 + bb0;
            out[row * 32 + c1] = v1 + bb1;
        }
    }
}

// ---------------------------------------------------------------------------
// Kernel B: hoisted norms. cet_norm[250] over 24 dims of cet_proj rows,
// code_norm[654] over the 8-dim code embeddings. max(norm, eps) pre-applied.
// ---------------------------------------------------------------------------
__global__ void norm_kernel(const float* __restrict__ cetp,
                            const float* __restrict__ code_emb,
                            float* __restrict__ cet_norm,
                            float* __restrict__ code_norm)
{
    int i = blockIdx.x * blockDim.x + threadIdx.x;
    if (i < 250) {
        const float* r = cetp + i * 32;
        float s = 0.f;
        #pragma unroll
        for (int j = 0; j < 24; ++j) s += r[j] * r[j];
        cet_norm[i] = fmaxf(sqrtf(s), EPSF);
    } else if (i < 250 + 654) {
        int k = i - 250;
        const float* r = code_emb + k * 8;
        float s = 0.f;
        #pragma unroll
        for (int j = 0; j < 8; ++j) s += r[j] * r[j];
        code_norm[k] = fmaxf(sqrtf(s), EPSF);
    }
}

// ---------------------------------------------------------------------------
// Kernel C: fused head. One 64-thread block per batch row b; thread t = candidate.
// Gathers, 4 cosine sims, 13->24->24->1 sigmoid MLP, softmax over C=64 in LDS.
// ---------------------------------------------------------------------------
__global__ __launch_bounds__(64)
void head_kernel(const float* __restrict__ xall, const float* __restrict__ xoth,
                 const float* __restrict__ xdoc, const float* __restrict__ cetp,
                 const float* __restrict__ cet_norm, const float* __restrict__ code_norm,
                 const float* __restrict__ code_emb,
                 const int* __restrict__ fcodes, const int* __restrict__ ccodes,
                 const float* __restrict__ gaz,
                 const float* __restrict__ W1, const float* __restrict__ B1,
                 const float* __restrict__ W2, const float* __restrict__ B2,
                 const float* __restrict__ Wl, const float* __restrict__ Bl,
                 float* __restrict__ out)
{
    __shared__ float sx[24], sxc[8], sxo[24], sxd[24];
    __shared__ float snorm[4];
    __shared__ float red[64];

    const int b = blockIdx.x;
    const int t = threadIdx.x;

    if (t < 24) {
        sx[t]  = xall[b * 32 + t];
        sxo[t] = xoth[b * 32 + t];
        sxd[t] = xdoc[b * 32 + t];
    } else if (t < 32) {
        sxc[t - 24] = xall[b * 32 + t];
    }
    __syncthreads();
    if (t < 4) {
        const float* v; int n;
        if      (t == 0) { v = sx;  n = 24; }
        else if (t == 1) { v = sxc; n = 8;  }
        else if (t == 2) { v = sxo; n = 24; }
        else             { v = sxd; n = 24; }
        float s = 0.f;
        for (int j = 0; j < n; ++j) s += v[j] * v[j];
        snorm[t] = fmaxf(sqrtf(s), EPSF);
    }
    __syncthreads();

    const int idx = b * 64 + t;
    const int cci = ccodes[idx];
    const int fci = fcodes[idx];

    const float* cet = cetp + cci * 32;
    float d1 = 0.f, d3 = 0.f, d4 = 0.f;
    #pragma unroll
    for (int j = 0; j < 24; ++j) {
        float cv = cet[j];
        d1 += sx[j] * cv; d3 += sxo[j] * cv; d4 += sxd[j] * cv;
    }
    const float* fe = code_emb + fci * 8;
    float d2 = 0.f;
    #pragma unroll
    for (int j = 0; j < 8; ++j) d2 += sxc[j] * fe[j];

    const float cn = cet_norm[cci];
    const float fn = code_norm[fci];

    float both[13];
    both[0] = d1 / (snorm[0] * cn);   // sim_country
    both[1] = d2 / (snorm[1] * fn);   // sim_code
    both[2] = d3 / (snorm[2] * cn);   // sim_other
    both[3] = d4 / (snorm[3] * cn);   // sim_doc
    const float* g = gaz + (size_t)idx * 9;
    #pragma unroll
    for (int j = 0; j < 9; ++j) both[4 + j] = g[j];

    float h1[24];
    #pragma unroll
    for (int m = 0; m < 24; ++m) {
        float a = B1[m];
        const float* wr = W1 + m * 13;
        #pragma unroll
        for (int j = 0; j < 13; ++j) a += wr[j] * both[j];
        h1[m] = 1.0f / (1.0f + expf(-a));
    }
    float h2[24];
    #pragma unroll
    for (int m = 0; m < 24; ++m) {
        float a = B2[m];
        const float* wr = W2 + m * 24;
        #pragma unroll
        for (int j = 0; j < 24; ++j) a += wr[j] * h1[j];
        h2[m] = 1.0f / (1.0f + expf(-a));
    }
    float last = Bl[0];
    #pragma unroll
    for (int m = 0; m < 24; ++m) last += Wl[m] * h2[m];

    // softmax over the 64 candidates of this row
    red[t] = last;
    __syncthreads();
    for (int s = 32; s > 0; s >>= 1) {
        if (t < s) red[t] = fmaxf(red[t], red[t + s]);
        __syncthreads();
    }
    float mx = red[0];
    __syncthreads();
    float e = expf(last - mx);
    red[t] = e;
    __syncthreads();
    for (int s = 32; s > 0; s >>= 1) {
        if (t < s) red[t] += red[t + s];
        __syncthreads();
    }
    out[idx] = e / red[0];
}

// ---------------------------------------------------------------------------
extern "C" void kernel_launch(void* const* d_in, const int* in_sizes, int n_in,
                              void* d_out, int out_size, void* d_ws, size_t ws_size,
                              hipStream_t stream)
{
    const float* placename = (const float*)d_in[0];
    const float* other     = (const float*)d_in[1];
    const float* doc       = (const float*)d_in[2];
    const int*   fcodes    = (const int*)d_in[3];
    const int*   ccodes    = (const int*)d_in[4];
    const float* gaz       = (const float*)d_in[5];
    const float* code_emb  = (const float*)d_in[6];
    const float* cemb      = (const float*)d_in[7];
    const float* W_cet  = (const float*)d_in[8];  const float* b_cet  = (const float*)d_in[9];
    const float* W_t2c  = (const float*)d_in[10]; const float* b_t2c  = (const float*)d_in[11];
    const float* W_ctx  = (const float*)d_in[12]; const float* b_ctx  = (const float*)d_in[13];
    const float* W_code = (const float*)d_in[14]; const float* b_code = (const float*)d_in[15];
    const float* W_mix1 = (const float*)d_in[16]; const float* b_mix1 = (const float*)d_in[17];
    const float* W_mix2 = (const float*)d_in[18]; const float* b_mix2 = (const float*)d_in[19];
    const float* W_last = (const float*)d_in[20]; const float* b_last = (const float*)d_in[21];

    float* ws        = (float*)d_ws;
    float* xall      = ws;                    // 2048*32
    float* xoth      = xall + 2048 * 32;      // 2048*32
    float* xdoc      = xoth + 2048 * 32;      // 2048*32
    float* cetp      = xdoc + 2048 * 32;      // 256*32 (250 valid rows)
    float* cet_norm  = cetp + 256 * 32;       // 256
    float* code_norm = cet_norm + 256;        // 654

    // jobs 0..2: 128 M-tiles each (2048 rows); job 3: 16 M-tiles (250 rows)
    proj_wmma_kernel<<<3 * 128 + 16, 256, 0, stream>>>(
        placename, other, doc, cemb,
        W_t2c, b_t2c, W_code, b_code, W_ctx, b_ctx, W_cet, b_cet,
        xall, xoth, xdoc, cetp);

    norm_kernel<<<4, 256, 0, stream>>>(cetp, code_emb, cet_norm, code_norm);

    head_kernel<<<2048, 64, 0, stream>>>(
        xall, xoth, xdoc, cetp, cet_norm, code_norm, code_emb,
        fcodes, ccodes, gaz,
        W_mix1, b_mix1, W_mix2, b_mix2, W_last, b_last,
        (float*)d_out);
}